// Model_29635274342607
// MI455X (gfx1250) — compile-verified
//
#include <hip/hip_runtime.h>
#include <hip/hip_bf16.h>
#include <math.h>

// ---------------------------------------------------------------------------
// VQ-VAE forward for MI455X (gfx1250, wave32).
// All stride==kernel (de)convolutions are lowered to implicit GEMMs executed
// with V_WMMA_F32_16X16X4_F32 (fp32 A/B, fp32 accum). GEMM configs are full
// compile-time template specializations so all address digit math folds to
// shifts/masks (no runtime integer division). The VQ nearest-neighbor search
// is a WMMA GEMM against an LDS-resident codebook with cross-lane argmin.
// ---------------------------------------------------------------------------

typedef __attribute__((ext_vector_type(2))) float v2f;
typedef __attribute__((ext_vector_type(8))) float v8f;

__device__ __forceinline__ v8f wmma_f32_k4(v2f a, v2f b, v8f c) {
  // D = A(16x4,f32) * B(4x16,f32) + C(16x16,f32)
  return __builtin_amdgcn_wmma_f32_16x16x4_f32(false, a, false, b, (short)0, c,
                                               false, false);
}

__device__ __forceinline__ float wave_sum(float v) {
#pragma unroll
  for (int o = 16; o > 0; o >>= 1) v += __shfl_xor(v, o, 32);
  return v;
}

constexpr int BG = 1 << 30;  // "infinite" modulus (power of two -> AND)

// --------------------------- generic conv-GEMM -----------------------------
// A[m][k]   = A[ base(m) + (k%KS)*SS + k/KS ]
// base(m)   = sum_i ((m/AD_i) % AM_i) * AS_i          (3 digits)
// Out index = obase(m) + (n/NDIV)*NSTR + (n%NDIV)     (obase: 3 digits)
// B[k][n]   = DECW ? W[k*N + (n%NCO)*(N/NCO) + n/NCO] : W[n*K + k]
template <int M, int K, int N, int NPAD, int KS, int SS,
          int AD1, int AM1, int AS1, int AD2, int AM2, int AS2,
          int AD3, int AM3, int AS3,
          int OD1, int OM1, int OS1, int OD2, int OM2, int OS2,
          int OD3, int OM3, int OS3,
          int NDIV, int NSTR, int NCO, int DECW>
__global__ __launch_bounds__(256) void gemm_conv_wmma(
    const float* __restrict__ A, const float* __restrict__ W,
    const float* __restrict__ bias, float* __restrict__ Out) {
  __shared__ float lb[K * 16];  // B slab: K x 16 (K <= 256)
  constexpr int KSPW = N / NCO;
  constexpr int NTILES = NPAD >> 4;
  const int tid = threadIdx.x;
  const int lane = tid & 31;
  const int wave = tid >> 5;
  const int m0 = (blockIdx.x * 8 + wave) * 16;
  const int row = m0 + (lane & 15);
  const int khalf = (lane >> 4) << 1;  // lanes 0-15 -> K 0/1, lanes 16-31 -> 2/3
  const int abase = ((row / AD1) % AM1) * AS1 + ((row / AD2) % AM2) * AS2 +
                    ((row / AD3) % AM3) * AS3;
  // pull this wave's A row toward the WGP before the WMMA loop
  __builtin_prefetch(A + abase, 0, 1);
#pragma unroll 1
  for (int nt = 0; nt < NTILES; ++nt) {
    __syncthreads();
    for (int i = tid; i < (K << 4); i += 256) {
      const int k = i >> 4;
      const int n = (nt << 4) + (i & 15);
      float v = 0.0f;
      if (n < N)
        v = DECW ? W[k * N + (n % NCO) * KSPW + n / NCO] : W[n * K + k];
      lb[i] = v;
    }
    __syncthreads();
    v8f acc = {0.f, 0.f, 0.f, 0.f, 0.f, 0.f, 0.f, 0.f};
    const int n16 = lane & 15;
    for (int k0 = 0; k0 < K; k0 += 4) {
      const int ka = k0 + khalf;
      const int kb = ka + 1;
      v2f a, b;
      a.x = A[abase + (ka % KS) * SS + ka / KS];
      a.y = A[abase + (kb % KS) * SS + kb / KS];
      b.x = lb[(ka << 4) + n16];
      b.y = lb[(kb << 4) + n16];
      acc = wmma_f32_k4(a, b, acc);
    }
    const int ncol = (nt << 4) + n16;
    if (ncol < N) {
      const float bv = bias[ncol % NCO];
      const int ooff = (ncol / NDIV) * NSTR + (ncol % NDIV);
      const int rbase = m0 + ((lane >> 4) << 3);  // C: upper half-lanes = +8
#pragma unroll
      for (int r = 0; r < 8; ++r) {
        const int orow = rbase + r;
        const int ob = ((orow / OD1) % OM1) * OS1 + ((orow / OD2) % OM2) * OS2 +
                       ((orow / OD3) % OM3) * OS3;
        Out[ob + ooff] = acc[r] + bv;
      }
    }
  }
}

// ------------------------- per-sample LayerNorm ----------------------------
template <int C>
__global__ __launch_bounds__(256) void ln_leaky(float* __restrict__ buf,
                                                const float* __restrict__ g,
                                                const float* __restrict__ beta,
                                                int S, float slope) {
  const int tid = threadIdx.x;
  float* p = buf + (size_t)blockIdx.x * S;
  float s = 0.f, s2 = 0.f;
  for (int i = tid; i < S; i += 256) {
    float v = p[i];
    s += v;
    s2 += v * v;
  }
  __shared__ float r0[8], r1[8];
  s = wave_sum(s);
  s2 = wave_sum(s2);
  if ((tid & 31) == 0) { r0[tid >> 5] = s; r1[tid >> 5] = s2; }
  __syncthreads();
  if (tid == 0) {
    float a = 0.f, b = 0.f;
    for (int i = 0; i < 8; ++i) { a += r0[i]; b += r1[i]; }
    r0[0] = a;
    r1[0] = b;
  }
  __syncthreads();
  const float n = (float)S;
  const float mean = r0[0] / n;
  const float var = (r1[0] - n * mean * mean) / (n - 1.0f);  // ddof=1
  const float inv = 1.0f / (sqrtf(var) + 1e-5f);             // (std + eps)
  for (int i = tid; i < S; i += 256) {
    const int ch = i & (C - 1);
    float y = g[ch] * ((p[i] - mean) * inv) + beta[ch];
    p[i] = y > 0.f ? y : slope * y;
  }
}

// ------------------------------ BatchNorm ----------------------------------
template <int C>
__global__ __launch_bounds__(256) void bn_stats(const float* __restrict__ x,
                                                float* __restrict__ stats,
                                                int M) {
  const int c = blockIdx.x;
  const int tid = threadIdx.x;
  float s = 0.f, s2 = 0.f;
  for (int m = tid; m < M; m += 256) {
    float v = x[(size_t)m * C + c];
    s += v;
    s2 += v * v;
  }
  __shared__ float r0[8], r1[8];
  s = wave_sum(s);
  s2 = wave_sum(s2);
  if ((tid & 31) == 0) { r0[tid >> 5] = s; r1[tid >> 5] = s2; }
  __syncthreads();
  if (tid == 0) {
    float a = 0.f, b = 0.f;
    for (int i = 0; i < 8; ++i) { a += r0[i]; b += r1[i]; }
    const float mean = a / (float)M;
    const float var = b / (float)M - mean * mean;  // biased var
    stats[c] = mean;
    stats[C + c] = rsqrtf(var + 1e-5f);
  }
}

template <int C>
__global__ __launch_bounds__(256) void bn_apply(float* __restrict__ x,
                                                const float* __restrict__ stats,
                                                const float* __restrict__ g,
                                                const float* __restrict__ beta,
                                                int total) {
  for (int i = blockIdx.x * 256 + threadIdx.x; i < total;
       i += gridDim.x * 256) {
    const int c = i & (C - 1);
    float v = (x[i] - stats[c]) * stats[C + c] * g[c] + beta[c];
    x[i] = v > 0.f ? v : 0.f;  // ReLU
  }
}

// ----------------------------- 8x8 MLP stack -------------------------------
struct MlpW {
  const float *w0, *b0, *w1, *b1, *w2, *b2, *w3, *b3;
};

__global__ __launch_bounds__(256) void mlp4(const float* __restrict__ in,
                                            float* __restrict__ out, MlpW mw,
                                            int rows, int act_last) {
  __shared__ float w[4][64];
  __shared__ float b[4][8];
  const int tid = threadIdx.x;
  if (tid < 64) {
    w[0][tid] = mw.w0[tid]; w[1][tid] = mw.w1[tid];
    w[2][tid] = mw.w2[tid]; w[3][tid] = mw.w3[tid];
  }
  if (tid < 8) {
    b[0][tid] = mw.b0[tid]; b[1][tid] = mw.b1[tid];
    b[2][tid] = mw.b2[tid]; b[3][tid] = mw.b3[tid];
  }
  __syncthreads();
  const int r = blockIdx.x * 256 + tid;
  if (r >= rows) return;
  float x[8], y[8];
#pragma unroll
  for (int i = 0; i < 8; ++i) x[i] = in[r * 8 + i];
  for (int l = 0; l < 4; ++l) {
#pragma unroll
    for (int j = 0; j < 8; ++j) {
      float acc = b[l][j];
#pragma unroll
      for (int i = 0; i < 8; ++i) acc += x[i] * w[l][j * 8 + i];
      y[j] = acc;
    }
    const bool act = act_last || (l < 3);
#pragma unroll
    for (int j = 0; j < 8; ++j)
      x[j] = act ? (y[j] > 0.f ? y[j] : 0.2f * y[j]) : y[j];
  }
#pragma unroll
  for (int i = 0; i < 8; ++i) out[r * 8 + i] = x[i];
}

// ------------------------------- VQ stage ----------------------------------
__global__ __launch_bounds__(256) void vq_norms(const float* __restrict__ E,
                                                float* __restrict__ snorm) {
  const int j = blockIdx.x * 256 + threadIdx.x;
  if (j < 2048) {
    float s = 0.f;
#pragma unroll
    for (int k = 0; k < 8; ++k) {
      float v = E[j * 8 + k];
      s += v * v;
    }
    snorm[j] = s;
  }
}

// argmin_j (|E_j|^2 - 2 fz.E_j) via WMMA; codebook in B-layout LDS (64KB).
__global__ __launch_bounds__(256) void vq_argmin(const float* __restrict__ fz,
                                                 const float* __restrict__ E,
                                                 const float* __restrict__ snorm,
                                                 int* __restrict__ idx) {
  extern __shared__ float Ec[];  // [k*2048 + j], 8*2048 floats
  const int tid = threadIdx.x;
  for (int i = tid; i < 2048 * 8; i += 256) {
    const int j = i & 2047;
    const int k = i >> 11;
    Ec[i] = E[j * 8 + k];
  }
  __syncthreads();
  const int lane = tid & 31;
  const int wave = tid >> 5;
  const int m0 = (blockIdx.x * 8 + wave) * 16;
  const int row = m0 + (lane & 15);
  const int khalf = (lane >> 4) << 1;
  v2f a0, a1;
  a0.x = fz[row * 8 + khalf];
  a0.y = fz[row * 8 + khalf + 1];
  a1.x = fz[row * 8 + 4 + khalf];
  a1.y = fz[row * 8 + 4 + khalf + 1];
  float minv[8];
  int mini[8];
#pragma unroll
  for (int r = 0; r < 8; ++r) { minv[r] = 3.4e38f; mini[r] = 0; }
#pragma unroll 2
  for (int jt = 0; jt < 128; ++jt) {
    const int j = (jt << 4) + (lane & 15);
    v2f b0, b1;
    b0.x = Ec[(khalf + 0) * 2048 + j];
    b0.y = Ec[(khalf + 1) * 2048 + j];
    b1.x = Ec[(khalf + 4) * 2048 + j];
    b1.y = Ec[(khalf + 5) * 2048 + j];
    v8f acc = {0.f, 0.f, 0.f, 0.f, 0.f, 0.f, 0.f, 0.f};
    acc = wmma_f32_k4(a0, b0, acc);
    acc = wmma_f32_k4(a1, b1, acc);
    const float sj = snorm[j];
#pragma unroll
    for (int r = 0; r < 8; ++r) {
      const float d = sj - 2.0f * acc[r];
      if (d < minv[r]) { minv[r] = d; mini[r] = j; }
    }
  }
#pragma unroll
  for (int off = 1; off < 16; off <<= 1) {
#pragma unroll
    for (int r = 0; r < 8; ++r) {
      const float ov = __shfl_xor(minv[r], off, 32);
      const int oi = __shfl_xor(mini[r], off, 32);
      if (ov < minv[r] || (ov == minv[r] && oi < mini[r])) {
        minv[r] = ov;
        mini[r] = oi;
      }
    }
  }
  if ((lane & 15) == 0) {
    const int rbase = m0 + ((lane >> 4) << 3);
#pragma unroll
    for (int r = 0; r < 8; ++r) idx[rbase + r] = mini[r];
  }
}

// gather q=E[idx], accumulate e-loss + histogram, run decoder 8x8 MLP stack
__global__ __launch_bounds__(256) void vq_gather_mlp(
    const float* __restrict__ fz, const float* __restrict__ E,
    const int* __restrict__ idx, MlpW mw, float* __restrict__ out,
    float* __restrict__ eacc, unsigned int* __restrict__ hist, int rows) {
  __shared__ float w[4][64];
  __shared__ float b[4][8];
  const int tid = threadIdx.x;
  if (tid < 64) {
    w[0][tid] = mw.w0[tid]; w[1][tid] = mw.w1[tid];
    w[2][tid] = mw.w2[tid]; w[3][tid] = mw.w3[tid];
  }
  if (tid < 8) {
    b[0][tid] = mw.b0[tid]; b[1][tid] = mw.b1[tid];
    b[2][tid] = mw.b2[tid]; b[3][tid] = mw.b3[tid];
  }
  __syncthreads();
  const int r = blockIdx.x * 256 + tid;
  float diff = 0.f;
  if (r < rows) {
    const int j = idx[r];
    atomicAdd(&hist[j], 1u);
    float x[8], y[8];
#pragma unroll
    for (int i = 0; i < 8; ++i) {
      const float q = E[j * 8 + i];
      const float dz = q - fz[r * 8 + i];
      diff += dz * dz;
      x[i] = q;  // straight-through: forward value is q
    }
    for (int l = 0; l < 4; ++l) {
#pragma unroll
      for (int jj = 0; jj < 8; ++jj) {
        float acc = b[l][jj];
#pragma unroll
        for (int i = 0; i < 8; ++i) acc += x[i] * w[l][jj * 8 + i];
        y[jj] = acc;
      }
#pragma unroll
      for (int jj = 0; jj < 8; ++jj)
        x[jj] = y[jj] > 0.f ? y[jj] : 0.2f * y[jj];  // act_last=True
    }
#pragma unroll
    for (int i = 0; i < 8; ++i) out[r * 8 + i] = x[i];
  }
  diff = wave_sum(diff);
  if ((tid & 31) == 0) atomicAdd(eacc, diff);
}

__global__ __launch_bounds__(256) void vq_finalize(
    const float* __restrict__ eacc, const unsigned int* __restrict__ hist,
    float* __restrict__ out, int out_last) {
  const int tid = threadIdx.x;
  float ent = 0.f;
  for (int i = tid; i < 2048; i += 256) {
    const float p = (float)hist[i] * (1.0f / 24576.0f);
    ent -= p * logf(p + 1e-10f);
  }
  __shared__ float r0[8];
  ent = wave_sum(ent);
  if ((tid & 31) == 0) r0[tid >> 5] = ent;
  __syncthreads();
  if (tid == 0) {
    float a = 0.f;
    for (int i = 0; i < 8; ++i) a += r0[i];
    out[0] = 0.25f * eacc[0] / (24576.0f * 8.0f);  // CC * e_loss
    out[out_last] = expf(a);                       // perplexity
  }
}

__global__ __launch_bounds__(256) void init_accum(unsigned int* hist,
                                                  float* eacc) {
  const int i = blockIdx.x * 256 + threadIdx.x;
  if (i < 2048) hist[i] = 0u;
  if (i == 0) eacc[0] = 0.f;
}

// ------------------------------- host side ---------------------------------
extern "C" void kernel_launch(void* const* d_in, const int* in_sizes, int n_in,
                              void* d_out, int out_size, void* d_ws,
                              size_t ws_size, hipStream_t stream) {
  (void)in_sizes; (void)n_in; (void)ws_size;
  auto f = [&](int i) { return (const float*)d_in[i]; };
  const float* x = f(0);
  const float* E = f(27);  // codebook

  // workspace layout (floats); decoder buffers alias consumed encoder buffers
  float* ws = (float*)d_ws;
  float* a1 = ws;                   // (B,4,16,18,16)  18,874,368
  float* a2 = a1 + 18874368;        // (B,4,4,18,32)    9,437,184
  float* a3 = a2 + 9437184;         // (B,4,4,6,64)     6,291,456
  float* a4 = a3 + 6291456;         // (B,4,1,6,128)    3,145,728
  float* he = a4 + 3145728;         // (B*24,8)
  float* fz = he + 196608;
  float* g1 = fz + 196608;
  int* vidx = (int*)(g1 + 196608);  // 24576
  float* stats = (float*)(vidx + 24576);              // 256
  unsigned int* hist = (unsigned int*)(stats + 256);  // 2048
  float* eacc = (float*)(hist + 2048);                // 1
  float* snorm = eacc + 1;                            // 2048
  float* d1 = a4;
  float* d2 = a3;
  float* d3 = a2;
  float* d4 = a1;
  float* out = (float*)d_out;
  float* xrec = out + 1;  // x_recon (B,64,72) right after loss scalar

  init_accum<<<8, 256, 0, stream>>>(hist, eacc);
  vq_norms<<<8, 256, 0, stream>>>(E, snorm);

  // ---------------- encoder ----------------
  // conv 1->16 k=(1,1,4)
  gemm_conv_wmma<1179648, 4, 16, 16, 4, 1, 18, BG, 72, 1, 18, 4, 1, 1, 0,
                 1, BG, 16, 1, 1, 0, 1, 1, 0, 16, 0, 16, 0>
      <<<1179648 / 128, 256, 0, stream>>>(x, f(1), f(2), a1);
  ln_leaky<16><<<1024, 256, 0, stream>>>(a1, f(3), f(4), 18432, 0.01f);
  // conv 16->32 k=(1,4,1)
  gemm_conv_wmma<294912, 64, 32, 32, 4, 288, 72, BG, 4608, 18, 4, 1152,
                 1, 18, 16, 1, BG, 32, 1, 1, 0, 1, 1, 0, 32, 0, 32, 0>
      <<<294912 / 128, 256, 0, stream>>>(a1, f(5), f(6), a2);
  ln_leaky<32><<<1024, 256, 0, stream>>>(a2, f(7), f(8), 9216, 0.01f);
  // conv 32->64 k=(1,1,3)
  gemm_conv_wmma<98304, 96, 64, 64, 3, 32, 6, BG, 576, 1, 6, 96, 1, 1, 0,
                 1, BG, 64, 1, 1, 0, 1, 1, 0, 64, 0, 64, 0>
      <<<98304 / 128, 256, 0, stream>>>(a2, f(9), f(10), a3);
  ln_leaky<64><<<1024, 256, 0, stream>>>(a3, f(11), f(12), 6144, 0.01f);
  // conv 64->128 k=(1,4,1)
  gemm_conv_wmma<24576, 256, 128, 128, 4, 384, 6, BG, 1536, 1, 6, 64, 1, 1, 0,
                 1, BG, 128, 1, 1, 0, 1, 1, 0, 128, 0, 128, 0>
      <<<24576 / 128, 256, 0, stream>>>(a3, f(13), f(14), a4);
  ln_leaky<128><<<1024, 256, 0, stream>>>(a4, f(15), f(16), 3072, 0.01f);
  // head 128->8 (1x1x1)
  gemm_conv_wmma<24576, 128, 8, 16, 1, 1, 1, BG, 128, 1, 1, 0, 1, 1, 0,
                 1, BG, 8, 1, 1, 0, 1, 1, 0, 8, 0, 8, 0>
      <<<24576 / 128, 256, 0, stream>>>(a4, f(17), f(18), he);
  MlpW em{f(19), f(20), f(21), f(22), f(23), f(24), f(25), f(26)};
  mlp4<<<96, 256, 0, stream>>>(he, fz, em, 24576, 0);

  // ---------------- VQ ----------------
  vq_argmin<<<192, 256, 65536, stream>>>(fz, E, snorm, vidx);
  MlpW dm{f(28), f(29), f(30), f(31), f(32), f(33), f(34), f(35)};
  vq_gather_mlp<<<96, 256, 0, stream>>>(fz, E, vidx, dm, g1, eacc, hist, 24576);
  vq_finalize<<<1, 256, 0, stream>>>(eacc, hist, out, out_size - 1);

  // ---------------- decoder ----------------
  // convT 8->128 (1x1x1)
  gemm_conv_wmma<24576, 8, 128, 128, 1, 1, 1, BG, 8, 1, 1, 0, 1, 1, 0,
                 1, BG, 128, 1, 1, 0, 1, 1, 0, 128, 0, 128, 1>
      <<<24576 / 128, 256, 0, stream>>>(g1, f(36), f(37), d1);
  bn_stats<128><<<128, 256, 0, stream>>>(d1, stats, 24576);
  bn_apply<128><<<2048, 256, 0, stream>>>(d1, stats, f(38), f(39), 3145728);
  // convT 128->64 k=(1,4,1)
  gemm_conv_wmma<24576, 128, 256, 256, 1, 1, 1, BG, 128, 1, 1, 0, 1, 1, 0,
                 6, BG, 1536, 1, 6, 64, 1, 1, 0, 64, 384, 64, 1>
      <<<24576 / 128, 256, 0, stream>>>(d1, f(40), f(41), d2);
  bn_stats<64><<<64, 256, 0, stream>>>(d2, stats, 98304);
  bn_apply<64><<<2048, 256, 0, stream>>>(d2, stats, f(42), f(43), 6291456);
  // convT 64->32 k=(1,1,3)
  gemm_conv_wmma<98304, 64, 96, 96, 1, 1, 1, BG, 64, 1, 1, 0, 1, 1, 0,
                 6, BG, 576, 1, 6, 96, 1, 1, 0, 96, 0, 32, 1>
      <<<98304 / 128, 256, 0, stream>>>(d2, f(44), f(45), d3);
  bn_stats<32><<<32, 256, 0, stream>>>(d3, stats, 294912);
  bn_apply<32><<<2048, 256, 0, stream>>>(d3, stats, f(46), f(47), 9437184);
  // convT 32->16 k=(1,4,1)
  gemm_conv_wmma<294912, 32, 64, 64, 1, 1, 1, BG, 32, 1, 1, 0, 1, 1, 0,
                 72, BG, 4608, 18, 4, 1152, 1, 18, 16, 16, 288, 16, 1>
      <<<294912 / 128, 256, 0, stream>>>(d3, f(48), f(49), d4);
  bn_stats<16><<<16, 256, 0, stream>>>(d4, stats, 1179648);
  bn_apply<16><<<2048, 256, 0, stream>>>(d4, stats, f(50), f(51), 18874368);
  // convT 16->1 k=(1,1,4) -> x_recon
  gemm_conv_wmma<1179648, 16, 4, 16, 1, 1, 1, BG, 16, 1, 1, 0, 1, 1, 0,
                 18, BG, 72, 1, 18, 4, 1, 1, 0, 4, 0, 1, 1>
      <<<1179648 / 128, 256, 0, stream>>>(d4, f(52), f(53), xrec);
}